// HungarianLoss_83365315215377
// MI455X (gfx1250) — compile-verified
//
#include <hip/hip_runtime.h>
#include <hip/hip_bf16.h>

#define NPTS 1024
#define NCH  256

typedef float v2f __attribute__((ext_vector_type(2)));
typedef float v8f __attribute__((ext_vector_type(8)));

// ---------------------------------------------------------------------------
// Phase 0: squared norms  x2[n] = sum_c X[n][c]^2   (X[n][c] = in[c*1024+n])
// ---------------------------------------------------------------------------
__global__ __launch_bounds__(256) void norms_kernel(
    const float* __restrict__ X, const float* __restrict__ Y,
    float* __restrict__ x2, float* __restrict__ y2)
{
    const int n = blockIdx.x * 256 + threadIdx.x;  // 0..1023, coalesced over n
    float sx = 0.0f, sy = 0.0f;
    for (int c = 0; c < NCH; ++c) {
        float a = X[c * NPTS + n];
        float b = Y[c * NPTS + n];
        sx = __builtin_fmaf(a, a, sx);
        sy = __builtin_fmaf(b, b, sy);
    }
    x2[n] = sx;
    y2[n] = sy;
}

// ---------------------------------------------------------------------------
// Phase 1: cost[i][j] = sqrt(max(x2[i] + y2[j] - 2*dot(X[i],Y[j]), 0))
// One wave computes one 16x16 tile with V_WMMA_F32_16X16X4_F32 (fp32 matrix
// pipe -> full precision for the downstream combinatorial solver).
// A (16x4 f32): lane = M (0..15), VGPR pair = K 2*(lane>>4)+{0,1}
// B (4x16 f32): lane = N, same K striping.  C/D: VGPR r -> M=r+8*(lane>>4).
// ---------------------------------------------------------------------------
__global__ __launch_bounds__(256) void cost_wmma_kernel(
    const float* __restrict__ X, const float* __restrict__ Y,
    const float* __restrict__ x2, const float* __restrict__ y2,
    float* __restrict__ cost)
{
    const int lane = threadIdx.x & 31;
    const int wave = threadIdx.x >> 5;
    const int wid  = blockIdx.x * 8 + wave;   // 0..4095 tiles (64x64)
    const int tm   = (wid >> 6) << 4;         // tile row base (M)
    const int tn   = (wid & 63) << 4;         // tile col base (N)
    const int lo   = lane & 15;
    const int hi   = lane >> 4;

    const int m = tm + lo;   // A: M = lane%16
    const int n = tn + lo;   // B: N = lane%16

    v8f acc = {};
    for (int kb = 0; kb < NCH; kb += 4) {
        const int k0 = kb + 2 * hi;
        v2f a, b;
        a.x = X[(k0    ) * NPTS + m];
        a.y = X[(k0 + 1) * NPTS + m];
        b.x = Y[(k0    ) * NPTS + n];
        b.y = Y[(k0 + 1) * NPTS + n];
        // D = A x B + C, fp32 matrix op, K = 4 per instruction
        acc = __builtin_amdgcn_wmma_f32_16x16x4_f32(
            /*neg_a=*/false, a, /*neg_b=*/false, b,
            /*c_mod=*/(short)0, acc, /*reuse_a=*/false, /*reuse_b=*/false);
    }

#pragma unroll
    for (int r = 0; r < 8; ++r) {
        const int M = tm + r + 8 * hi;
        const int N = tn + lo;
        float d2 = x2[M] + y2[N] - 2.0f * acc[r];
        cost[M * NPTS + N] = __builtin_sqrtf(fmaxf(d2, 0.0f));
    }
}

// ---------------------------------------------------------------------------
// Phase 2: Jonker-Volgenant shortest augmenting path LAP.
// One 1024-thread block (32 wave32 waves). Thread j owns column j:
// v[j], shortest[j], SC[j] are registers. u/row4col/col4row/path in LDS.
// Masked argmin: monotone-float key packed with index into u64, wave32
// shfl_xor reduce + one cross-wave LDS stage. Cost rows hit the 192MB L2.
// ---------------------------------------------------------------------------
__global__ __launch_bounds__(1024) void lap_kernel(
    const float* __restrict__ cost, int* __restrict__ g_col4row)
{
    __shared__ float u[NPTS];
    __shared__ int   row4col[NPTS];
    __shared__ int   col4row[NPTS];
    __shared__ int   pathj[NPTS];
    __shared__ unsigned long long red[32];
    __shared__ float s_minVal;
    __shared__ int   s_i, s_sink, s_jstar;

    const int j    = threadIdx.x;
    const int lane = j & 31;
    const int wv   = j >> 5;
    const float INF = __builtin_inff();

    float vj = 0.0f;                // dual v[j] (register)
    u[j] = 0.0f;
    row4col[j] = -1;
    col4row[j] = -1;
    __syncthreads();

    for (int cur = 0; cur < NPTS; ++cur) {
        float shortest = INF;       // shortest[j] (register)
        bool  sc = false;           // SC[j]      (register)
        pathj[j] = 0;
        if (j == 0) { s_i = cur; s_minVal = 0.0f; s_sink = -1; }
        __syncthreads();

        while (s_sink < 0) {
            const int   i  = s_i;
            const float mv = s_minVal;
            const float ui = u[i];          // LDS broadcast read
            if (!sc) {
                float r = mv + cost[i * NPTS + j] - ui - vj;  // L2-resident row
                if (r < shortest) { shortest = r; pathj[j] = i; }
            }
            // ---- block argmin over (sc ? INF : shortest, j) ----
            float masked = sc ? INF : shortest;
            unsigned kb = __float_as_uint(masked);
            kb = (kb & 0x80000000u) ? ~kb : (kb | 0x80000000u);  // order-preserving
            unsigned long long pk =
                ((unsigned long long)kb << 32) | (unsigned)j;
#pragma unroll
            for (int o = 16; o > 0; o >>= 1) {
                unsigned long long other = __shfl_xor(pk, o, 32);
                if (other < pk) pk = other;
            }
            if (lane == 0) red[wv] = pk;
            __syncthreads();
            if (wv == 0) {
                unsigned long long p2 = red[lane];
#pragma unroll
                for (int o = 16; o > 0; o >>= 1) {
                    unsigned long long other = __shfl_xor(p2, o, 32);
                    if (other < p2) p2 = other;
                }
                if (lane == 0) s_jstar = (int)(p2 & 0xFFFFFFFFu);
            }
            __syncthreads();
            if (j == s_jstar) {             // winner thread advances the search
                sc = true;
                s_minVal = shortest;
                int rc = row4col[j];
                if (rc < 0) s_sink = j;
                else        s_i    = rc;
            }
            __syncthreads();
        }

        // ---- dual updates (row4col is injective -> conflict-free scatter) ----
        const float mv = s_minVal;
        float delta = mv - shortest;
        if (j == 0) u[cur] += mv;
        if (sc) {
            int rc = row4col[j];
            if (rc >= 0) u[rc] += delta;    // rc != cur (cur unassigned)
            vj -= delta;
        }
        __syncthreads();

        // ---- augment along alternating path (serial, thread 0) ----
        if (j == 0) {
            int jj = s_sink;
            for (;;) {
                int ii = pathj[jj];
                row4col[jj] = ii;
                int prev = col4row[ii];
                col4row[ii] = jj;
                if (ii == cur) break;
                jj = prev;
            }
        }
        __syncthreads();
    }

    g_col4row[j] = col4row[j];
}

// ---------------------------------------------------------------------------
// Phase 3: loss = mean((X - Y[col])^2) over 1024x256 elements -> scalar f32
// ---------------------------------------------------------------------------
__global__ __launch_bounds__(1024) void loss_kernel(
    const float* __restrict__ X, const float* __restrict__ Y,
    const int* __restrict__ col, float* __restrict__ out)
{
    __shared__ float red[32];
    const int i  = threadIdx.x;       // row
    const int ci = col[i];            // assigned column
    float s = 0.0f;
    for (int c = 0; c < NCH; ++c) {
        float d = X[c * NPTS + i] - Y[c * NPTS + ci];
        s = __builtin_fmaf(d, d, s);
    }
#pragma unroll
    for (int o = 16; o > 0; o >>= 1) s += __shfl_xor(s, o, 32);
    if ((i & 31) == 0) red[i >> 5] = s;
    __syncthreads();
    if (i < 32) {
        float t = red[i];
#pragma unroll
        for (int o = 16; o > 0; o >>= 1) t += __shfl_xor(t, o, 32);
        if (i == 0) out[0] = t * (1.0f / (float)(NPTS * NCH));
    }
}

// ---------------------------------------------------------------------------
extern "C" void kernel_launch(void* const* d_in, const int* in_sizes, int n_in,
                              void* d_out, int out_size, void* d_ws, size_t ws_size,
                              hipStream_t stream)
{
    (void)in_sizes; (void)n_in; (void)out_size; (void)ws_size;
    const float* X = (const float*)d_in[0];   // input  (1,256,32,32) fp32
    const float* Y = (const float*)d_in[1];   // target (1,256,32,32) fp32

    // workspace layout
    float* cost = (float*)d_ws;                                   // 4 MB
    float* x2   = (float*)((char*)d_ws + (size_t)NPTS * NPTS * 4);// 4 KB
    float* y2   = x2 + NPTS;                                      // 4 KB
    int*   col  = (int*)(y2 + NPTS);                              // 4 KB
    float* out  = (float*)d_out;

    norms_kernel    <<<NPTS / 256, 256, 0, stream>>>(X, Y, x2, y2);
    cost_wmma_kernel<<<(NPTS / 16) * (NPTS / 16) / 8, 256, 0, stream>>>(X, Y, x2, y2, cost);
    lap_kernel      <<<1, NPTS, 0, stream>>>(cost, col);
    loss_kernel     <<<1, NPTS, 0, stream>>>(X, Y, col, out);
}